// GCN_36661840838722
// MI455X (gfx1250) — compile-verified
//
#include <hip/hip_runtime.h>
#include <hip/hip_bf16.h>

typedef __attribute__((ext_vector_type(2))) float v2f;
typedef __attribute__((ext_vector_type(8))) float v8f;

#define DH 128   // hidden dim (all layer GEMMs are K=128 -> N=128)

// ---------------------------------------------------------------------------
// Degree / normalization prep
// ---------------------------------------------------------------------------
__global__ void deg_init(float* __restrict__ deg, int n) {
    int t = blockIdx.x * blockDim.x + threadIdx.x;
    if (t < n) deg[t] = 1.0f;              // self-loop contributes 1
}

__global__ void deg_accum(const long long* __restrict__ ei, float* __restrict__ deg, int E) {
    int t = blockIdx.x * blockDim.x + threadIdx.x;
    if (t < E) atomicAdd(&deg[(int)ei[(size_t)E + t]], 1.0f);   // dst row of edge_index
}

__global__ void deg_rsqrt(float* __restrict__ dis, int n) {
    int t = blockIdx.x * blockDim.x + threadIdx.x;
    if (t < n) dis[t] = rsqrtf(dis[t]);    // deg >= 1 always (self-loops)
}

__global__ void edge_prep(const long long* __restrict__ ei, const float* __restrict__ dis,
                          int* __restrict__ srcI, int* __restrict__ dstI,
                          float* __restrict__ normE, int E) {
    int t = blockIdx.x * blockDim.x + threadIdx.x;
    if (t >= E) return;
    int s = (int)ei[t];
    int d = (int)ei[(size_t)E + t];
    srcI[t] = s;
    dstI[t] = d;
    normE[t] = dis[s] * dis[d];
}

// ---------------------------------------------------------------------------
// WMMA fp32 GEMM: HW = H(16-row block) @ W (128x128), plus AGG = HW * dis^2
// One wave (32 threads) per 16x128 output block; 8 accumulator tiles.
// A/B fragment layouts per CDNA5 ISA 7.12.2 (f32 16x16x4).
// ---------------------------------------------------------------------------
__global__ void __launch_bounds__(32)
gcn_gemm(const float* __restrict__ H, const float* __restrict__ W,
         const float* __restrict__ dis, float* __restrict__ HW,
         float* __restrict__ AGG, int nrows) {
    __shared__ float sA[16][132];          // 132-pad -> conflict-free b64 reads
    int row0 = blockIdx.x * 16;
    if (row0 >= nrows) return;

    // Stage 16x128 A tile into LDS, coalesced float4.
    const float4* src4 = (const float4*)(H + (size_t)row0 * DH);
    for (int i = threadIdx.x; i < 16 * 32; i += 32) {
        int r = i >> 5, c4 = i & 31;
        *(float4*)&sA[r][c4 * 4] = src4[r * 32 + c4];
    }
    __syncthreads();

    int lane = threadIdx.x;
    int lr = lane & 15;                    // M (A/D col-lane) / N (B col)
    int hi = lane >> 4;                    // K-half selector

    v8f acc[8] = {};
    for (int kb = 0; kb < 32; ++kb) {      // K = 128 in steps of 4
        int K0 = kb * 4;
        v2f a;
        a.x = sA[lr][K0 + 2 * hi];         // A: v0 = K0|K0+2, v1 = K0+1|K0+3
        a.y = sA[lr][K0 + 2 * hi + 1];
        #pragma unroll
        for (int t = 0; t < 8; ++t) {      // 8 column tiles of W
            v2f b;
            b.x = W[(size_t)(K0 + 2 * hi) * DH + t * 16 + lr];
            b.y = W[(size_t)(K0 + 2 * hi + 1) * DH + t * 16 + lr];
            acc[t] = __builtin_amdgcn_wmma_f32_16x16x4_f32(
                false, a, false, b, (short)0, acc[t], false, false);
        }
    }

    // Self-loop scale factors for the 8 rows this half-wave owns.
    float dis2[8];
    #pragma unroll
    for (int j = 0; j < 8; ++j) {
        float d = dis[row0 + j + 8 * hi];
        dis2[j] = d * d;
    }

    // D layout: VGPR j -> rows (j | j+8), N = lane%16.
    #pragma unroll
    for (int t = 0; t < 8; ++t) {
        #pragma unroll
        for (int j = 0; j < 8; ++j) {
            int r = row0 + j + 8 * hi;
            float v = acc[t][j];
            size_t o = (size_t)r * DH + t * 16 + lr;
            HW[o] = v;
            AGG[o] = v * dis2[j];          // init aggregation with self-loop term
        }
    }
}

// ---------------------------------------------------------------------------
// Edge scatter: one wave per edge, each lane handles 4 channels (float4).
// agg[dst] += hw[src] * norm   (atomic f32, no-return form)
// ---------------------------------------------------------------------------
__global__ void gcn_scatter(const float* __restrict__ HW, float* __restrict__ AGG,
                            const int* __restrict__ srcI, const int* __restrict__ dstI,
                            const float* __restrict__ normE, int E) {
    int t = blockIdx.x * blockDim.x + threadIdx.x;
    int e = t >> 5;
    if (e >= E) return;
    int q = t & 31;
    int s = srcI[e];                       // wave-uniform -> scalarized loads
    int d = dstI[e];
    float w = normE[e];
    float4 v = ((const float4*)(HW + (size_t)s * DH))[q];
    float* ap = AGG + (size_t)d * DH + q * 4;
    atomicAdd(ap + 0, v.x * w);
    atomicAdd(ap + 1, v.y * w);
    atomicAdd(ap + 2, v.z * w);
    atomicAdd(ap + 3, v.w * w);
}

// ---------------------------------------------------------------------------
// h = relu(agg + bias)
// ---------------------------------------------------------------------------
__global__ void gcn_finalize(const float* __restrict__ AGG, const float* __restrict__ bias,
                             float* __restrict__ Hout, int n) {
    int t = blockIdx.x * blockDim.x + threadIdx.x;
    int node = t >> 5;
    if (node >= n) return;
    int q = t & 31;
    float4 v = ((const float4*)(AGG + (size_t)node * DH))[q];
    float4 b = ((const float4*)bias)[q];
    v.x = fmaxf(v.x + b.x, 0.0f);
    v.y = fmaxf(v.y + b.y, 0.0f);
    v.z = fmaxf(v.z + b.z, 0.0f);
    v.w = fmaxf(v.w + b.w, 0.0f);
    ((float4*)(Hout + (size_t)node * DH))[q] = v;
}

// ---------------------------------------------------------------------------
// Mean pool + FC head
// ---------------------------------------------------------------------------
__global__ void zero_vec(float* __restrict__ a, int n) {
    int t = blockIdx.x * blockDim.x + threadIdx.x;
    if (t < n) a[t] = 0.0f;
}

__global__ void pool_sum(const float* __restrict__ H, float* __restrict__ accum, int n) {
    int c = threadIdx.x;                   // 128 threads = one column each
    float s = 0.0f;
    for (int r = blockIdx.x; r < n; r += gridDim.x)
        s += H[(size_t)r * DH + c];
    atomicAdd(&accum[c], s);
}

__global__ void fc_out(const float* __restrict__ accum, const float* __restrict__ Wfc,
                       const float* __restrict__ bfc, float* __restrict__ out, float invN) {
    int o = threadIdx.x;
    if (o >= 32) return;
    float s = bfc[o];
    #pragma unroll 4
    for (int k = 0; k < DH; ++k)
        s += accum[k] * invN * Wfc[k * 32 + o];
    out[o] = s;
}

// ---------------------------------------------------------------------------
// Launch
// ---------------------------------------------------------------------------
extern "C" void kernel_launch(void* const* d_in, const int* in_sizes, int n_in,
                              void* d_out, int out_size, void* d_ws, size_t ws_size,
                              hipStream_t stream) {
    const float*     x   = (const float*)d_in[0];
    const long long* ei  = (const long long*)d_in[1];
    const float*     W0  = (const float*)d_in[2];
    const float*     b0  = (const float*)d_in[3];
    const float*     W1  = (const float*)d_in[4];
    const float*     b1  = (const float*)d_in[5];
    const float*     W2  = (const float*)d_in[6];
    const float*     b2  = (const float*)d_in[7];
    const float*     Wfc = (const float*)d_in[8];
    const float*     bfc = (const float*)d_in[9];

    const int N = in_sizes[0] / DH;        // 100000
    const int E = in_sizes[1] / 2;         // 1600000

    // Workspace carve (256B aligned slices)
    auto align_up = [](size_t v) { return (v + 255) & ~(size_t)255; };
    char* p = (char*)d_ws;
    float* dis   = (float*)p;                 p += align_up((size_t)N * 4);
    float* normE = (float*)p;                 p += align_up((size_t)E * 4);
    int*   srcI  = (int*)p;                   p += align_up((size_t)E * 4);
    int*   dstI  = (int*)p;                   p += align_up((size_t)E * 4);
    float* bufH  = (float*)p;                 p += align_up((size_t)N * DH * 4);
    float* bufHW = (float*)p;                 p += align_up((size_t)N * DH * 4);
    float* bufAG = (float*)p;                 p += align_up((size_t)N * DH * 4);
    float* accum = (float*)p;                 p += align_up(DH * 4);
    (void)ws_size; (void)n_in; (void)out_size;

    // 1. Degree + symmetric normalization (computed once, reused for 3 layers)
    deg_init <<<(N + 255) / 256, 256, 0, stream>>>(dis, N);
    deg_accum<<<(E + 255) / 256, 256, 0, stream>>>(ei, dis, E);
    deg_rsqrt<<<(N + 255) / 256, 256, 0, stream>>>(dis, N);
    edge_prep<<<(E + 255) / 256, 256, 0, stream>>>(ei, dis, srcI, dstI, normE, E);

    // 2. Three GCN layers
    const float* Ws[3] = {W0, W1, W2};
    const float* bs[3] = {b0, b1, b2};
    const float* hin = x;
    const int gemm_grid = (N + 15) / 16;
    const int scat_grid = (int)(((size_t)E * 32 + 255) / 256);
    const int fin_grid  = (int)(((size_t)N * 32 + 255) / 256);
    for (int l = 0; l < 3; ++l) {
        gcn_gemm    <<<gemm_grid, 32, 0, stream>>>(hin, Ws[l], dis, bufHW, bufAG, N);
        gcn_scatter <<<scat_grid, 256, 0, stream>>>(bufHW, bufAG, srcI, dstI, normE, E);
        gcn_finalize<<<fin_grid, 256, 0, stream>>>(bufAG, bs[l], bufH, N);
        hin = bufH;
    }

    // 3. Mean pool + FC
    zero_vec<<<1, 128, 0, stream>>>(accum, DH);
    pool_sum<<<256, 128, 0, stream>>>(bufH, accum, N);
    fc_out  <<<1, 32, 0, stream>>>(accum, Wfc, bfc, (float*)d_out, 1.0f / (float)N);
}